// DownSampler_46420006535685
// MI455X (gfx1250) — compile-verified
//
#include <hip/hip_runtime.h>
#include <hip/hip_bf16.h>

typedef float v2f __attribute__((ext_vector_type(2)));
typedef float v8f __attribute__((ext_vector_type(8)));

#define RS_BLOCK 256
#define RS_ELEMS 8
#define RS_CHUNK (RS_BLOCK * RS_ELEMS)  // 2048 elements per block per pass
#define D 128
#define GEMM_ROWS 64                    // M rows per block (4 x 16-row tiles)

// ---------------------------------------------------------------------------
// Stage 1: per-row squared L2 norm -> monotonic sortable u32 key + index.
// One wave32 per row: 32 lanes x float4 = 128 floats.
// ---------------------------------------------------------------------------
__global__ void score_keys_kernel(const float* __restrict__ x,
                                  unsigned* __restrict__ keys,
                                  unsigned* __restrict__ vals, int n) {
  int row = blockIdx.x * (blockDim.x >> 5) + (threadIdx.x >> 5);
  int lane = threadIdx.x & 31;
  if (row >= n) return;
  const float4* r = (const float4*)(x + (size_t)row * D);
  float4 v = r[lane];
  float s = v.x * v.x + v.y * v.y + v.z * v.z + v.w * v.w;
#pragma unroll
  for (int off = 16; off > 0; off >>= 1) s += __shfl_xor(s, off, 32);
  if (lane == 0) {
    unsigned u = __float_as_uint(s);
    u = (u & 0x80000000u) ? ~u : (u | 0x80000000u);  // monotonic float->uint
    keys[row] = u;
    vals[row] = (unsigned)row;
  }
}

// ---------------------------------------------------------------------------
// Stage 2: stable LSD radix sort, 8 passes of 4 bits (wave32 ballot ranking).
// ---------------------------------------------------------------------------
__global__ void radix_count_kernel(const unsigned* __restrict__ keys,
                                   unsigned* __restrict__ counts, int n,
                                   int shift, int numChunks) {
  __shared__ unsigned hist[16];
  int t = threadIdx.x;
  if (t < 16) hist[t] = 0;
  __syncthreads();
  int base = blockIdx.x * RS_CHUNK;
#pragma unroll
  for (int e = 0; e < RS_ELEMS; ++e) {
    int idx = base + e * RS_BLOCK + t;
    if (idx < n) atomicAdd(&hist[(keys[idx] >> shift) & 15u], 1u);
  }
  __syncthreads();
  if (t < 16) counts[t * numChunks + blockIdx.x] = hist[t];
}

__global__ void radix_scan_kernel(unsigned* __restrict__ counts, int L) {
  __shared__ unsigned sums[256];
  int t = threadIdx.x;
  int per = (L + 255) / 256;
  int s0 = t * per;
  int s1 = s0 + per;
  if (s1 > L) s1 = L;
  unsigned s = 0;
  for (int i = s0; i < s1; ++i) s += counts[i];
  sums[t] = s;
  __syncthreads();
  for (int off = 1; off < 256; off <<= 1) {
    unsigned v = (t >= off) ? sums[t - off] : 0u;
    __syncthreads();
    sums[t] += v;
    __syncthreads();
  }
  unsigned run = (t == 0) ? 0u : sums[t - 1];
  for (int i = s0; i < s1; ++i) {
    unsigned c = counts[i];
    counts[i] = run;
    run += c;
  }
}

__global__ void radix_scatter_kernel(const unsigned* __restrict__ keysIn,
                                     const unsigned* __restrict__ valsIn,
                                     unsigned* __restrict__ keysOut,
                                     unsigned* __restrict__ valsOut,
                                     const unsigned* __restrict__ offsets,
                                     int n, int shift, int numChunks) {
  __shared__ unsigned digitOff[16];
  __shared__ unsigned waveHist[8][16];  // 8 wave32s per 256-thread block
  int t = threadIdx.x;
  int lane = t & 31;
  int wave = t >> 5;
  if (t < 16) digitOff[t] = offsets[t * numChunks + blockIdx.x];
  int base = blockIdx.x * RS_CHUNK;
#pragma unroll 1
  for (int e = 0; e < RS_ELEMS; ++e) {
    if (t < 128) ((unsigned*)waveHist)[t] = 0;
    __syncthreads();
    int idx = base + e * RS_BLOCK + t;
    bool valid = idx < n;
    unsigned key = valid ? keysIn[idx] : 0u;
    unsigned val = valid ? valsIn[idx] : 0u;
    unsigned digit = (key >> shift) & 15u;
    // wave32 same-digit mask via 4 ballots
    unsigned mask = (unsigned)__ballot(valid);
#pragma unroll
    for (int b = 0; b < 4; ++b) {
      unsigned bal = (unsigned)__ballot((int)((digit >> b) & 1u));
      mask &= ((digit >> b) & 1u) ? bal : ~bal;
    }
    unsigned rank = __popc(mask & ((1u << lane) - 1u));
    if (valid && rank == 0) waveHist[wave][digit] = __popc(mask);
    __syncthreads();
    if (valid) {
      unsigned woff = 0;
      for (int w = 0; w < wave; ++w) woff += waveHist[w][digit];
      unsigned pos = digitOff[digit] + woff + rank;
      keysOut[pos] = key;
      valsOut[pos] = val;
    }
    __syncthreads();
    if (t < 16) {
      unsigned tot = 0;
#pragma unroll
      for (int w = 0; w < 8; ++w) tot += waveHist[w][t];
      digitOff[t] += tot;
    }
    __syncthreads();
  }
}

// ---------------------------------------------------------------------------
// Stage 3a: gather pos (3 floats) and batch (raw i32) for the kept nodes.
// ---------------------------------------------------------------------------
__global__ void gather_aux_kernel(const float* __restrict__ pos,
                                  const int* __restrict__ batch,
                                  const unsigned* __restrict__ topVals,
                                  float* __restrict__ posOut,
                                  int* __restrict__ batchOut, int keep) {
  int j = blockIdx.x * blockDim.x + threadIdx.x;
  if (j >= keep) return;
  unsigned p = topVals[j];
  posOut[(size_t)j * 3 + 0] = pos[(size_t)p * 3 + 0];
  posOut[(size_t)j * 3 + 1] = pos[(size_t)p * 3 + 1];
  posOut[(size_t)j * 3 + 2] = pos[(size_t)p * 3 + 2];
  batchOut[j] = batch[p];
}

// ---------------------------------------------------------------------------
// Stage 3b: gathered GEMM  out[j,:] = x[perm[j],:] @ W + b  via
// V_WMMA_F32_16X16X4_F32. Block = 256 threads = 8 waves; each block does a
// 64-row output stripe (4 x 16-row M tiles); wave w owns output columns
// [16w, 16w+16). The wave's 128x16 W slab (64 VGPRs/lane) is loaded ONCE per
// block and reused across all 4 M tiles, so the steady-state inner loop is
// pure ds_load + v_wmma (no global traffic).
// ---------------------------------------------------------------------------
__global__ void topk_gemm_kernel(const float* __restrict__ x,
                                 const float* __restrict__ W,
                                 const float* __restrict__ bias,
                                 const unsigned* __restrict__ topVals,
                                 float* __restrict__ out, int keep) {
  __shared__ float tileA[GEMM_ROWS][D + 4];  // +4 pad -> conflict-free rows
  int t = threadIdx.x;
  int lane = t & 31;
  int wave = t >> 5;
  int rowBase = blockIdx.x * GEMM_ROWS;

  int n0 = wave * 16;        // this wave's output column base
  int nc = lane & 15;        // N = lane % 16 (B and C/D layouts)
  int kHi = (lane >> 4) * 2; // lanes 16..31 hold K=2,3 of each K-quad
  int mHi = (lane >> 4) * 8; // C/D: VGPR r -> M = r (+8 for upper half-wave)

  // Load this wave's B slab (W columns [n0, n0+16), all K) into registers:
  // 32 K-quads x v2f = 64 VGPRs per lane, fetched once per block.
  v2f breg[D / 4];
#pragma unroll
  for (int i = 0; i < D / 4; ++i) {
    breg[i][0] = W[(size_t)(4 * i + kHi + 0) * D + n0 + nc];
    breg[i][1] = W[(size_t)(4 * i + kHi + 1) * D + n0 + nc];
  }
  float bb = bias[n0 + nc];

  // Stage 64 gathered rows of x into LDS (each thread moves 8x float4).
  for (int i = t; i < GEMM_ROWS * (D / 4); i += 256) {
    int r = i >> 5;          // row 0..63
    int c4 = i & 31;         // float4 column 0..31
    int gr = rowBase + r;
    unsigned src = topVals[(gr < keep) ? gr : 0];  // clamp for tail tile
    float4 v = ((const float4*)(x + (size_t)src * D))[c4];
    tileA[r][c4 * 4 + 0] = v.x;
    tileA[r][c4 * 4 + 1] = v.y;
    tileA[r][c4 * 4 + 2] = v.z;
    tileA[r][c4 * 4 + 3] = v.w;
  }
  __syncthreads();

#pragma unroll 1
  for (int tile = 0; tile < GEMM_ROWS / 16; ++tile) {
    int mA = tile * 16 + nc;  // A: M = lane&15 within this tile

    v8f acc;
#pragma unroll
    for (int r = 0; r < 8; ++r) acc[r] = bb;  // seed C with bias: D = A*B + C

#pragma unroll
    for (int i = 0; i < D / 4; ++i) {
      v2f a;
      a[0] = tileA[mA][4 * i + kHi + 0];
      a[1] = tileA[mA][4 * i + kHi + 1];
      acc = __builtin_amdgcn_wmma_f32_16x16x4_f32(
          /*neg_a=*/false, a, /*neg_b=*/false, breg[i],
          /*c_mod=*/(short)0, acc, /*reuse_a=*/false, /*reuse_b=*/false);
    }

#pragma unroll
    for (int r = 0; r < 8; ++r) {
      int m = rowBase + tile * 16 + mHi + r;
      if (m < keep) out[(size_t)m * D + n0 + nc] = acc[r];
    }
  }
}

// ---------------------------------------------------------------------------
extern "C" void kernel_launch(void* const* d_in, const int* in_sizes, int n_in,
                              void* d_out, int out_size, void* d_ws, size_t ws_size,
                              hipStream_t stream) {
  const float* x   = (const float*)d_in[0];
  const float* pos = (const float*)d_in[1];
  const int* batch = (const int*)d_in[2];
  const float* W   = (const float*)d_in[3];
  const float* b   = (const float*)d_in[4];

  int n = in_sizes[2];                 // number of nodes
  int keep = n / 4;                    // RATIO = 0.25, int() truncation
  if (keep < 1) keep = 1;

  float* outX = (float*)d_out;                       // [keep, 128]
  float* outPos = outX + (size_t)keep * D;           // [keep, 3]
  int* outBatch = (int*)(outPos + (size_t)keep * 3); // [keep] raw i32 bits

  unsigned* keys0 = (unsigned*)d_ws;
  unsigned* vals0 = keys0 + n;
  unsigned* keys1 = vals0 + n;
  unsigned* vals1 = keys1 + n;
  unsigned* counts = vals1 + n;

  int numChunks = (n + RS_CHUNK - 1) / RS_CHUNK;

  // 1) scores -> sortable keys
  int rowsPerBlock = 256 / 32;
  score_keys_kernel<<<(n + rowsPerBlock - 1) / rowsPerBlock, 256, 0, stream>>>(
      x, keys0, vals0, n);

  // 2) stable LSD radix sort ascending (8 x 4-bit passes; ends in keys0/vals0)
  unsigned *ka = keys0, *va = vals0, *kb = keys1, *vb = vals1;
  for (int p = 0; p < 8; ++p) {
    int shift = p * 4;
    radix_count_kernel<<<numChunks, RS_BLOCK, 0, stream>>>(ka, counts, n, shift,
                                                           numChunks);
    radix_scan_kernel<<<1, 256, 0, stream>>>(counts, 16 * numChunks);
    radix_scatter_kernel<<<numChunks, RS_BLOCK, 0, stream>>>(
        ka, va, kb, vb, counts, n, shift, numChunks);
    unsigned* tk = ka; ka = kb; kb = tk;
    unsigned* tv = va; va = vb; vb = tv;
  }
  const unsigned* topVals = va + (n - keep);  // top-keep indices, ascending

  // 3) gathers + WMMA GEMM
  gather_aux_kernel<<<(keep + 255) / 256, 256, 0, stream>>>(
      pos, batch, topVals, outPos, outBatch, keep);
  topk_gemm_kernel<<<(keep + GEMM_ROWS - 1) / GEMM_ROWS, 256, 0, stream>>>(
      x, W, b, topVals, outX, keep);
}